// GConvGRUManual_86827058856609
// MI455X (gfx1250) — compile-verified
//
#include <hip/hip_runtime.h>
#include <math.h>

// ---------------------------------------------------------------------------
// GConvGRU (GCN-gated GRU cell) for MI455X / gfx1250.
// Memory-bound: 3 edge scatter passes dominate (~5-7GB @ 23.3TB/s).
// GEMMs (39 GFLOP total) use native fp32 WMMA (V_WMMA_F32_16X16X4_F32) —
// compute headroom is enormous, fp32 keeps reference accuracy.
// A-strip staging uses gfx1250 GLOBAL_LOAD_ASYNC_TO_LDS_B128 (ASYNCcnt).
// ---------------------------------------------------------------------------

#define HIDC 256        // hidden / input channels
#define DINC 512        // concat width
#define LDS_STRIDE 516  // 512 + 4 pad: rows 16B-aligned for async B128,
                        // 516 % 64(banks) = 4 -> conflict-free ds_load_b64

typedef float v2f __attribute__((ext_vector_type(2)));
typedef float v8f __attribute__((ext_vector_type(8)));

__device__ __forceinline__ void atomic_add_f32(float* p, float v) {
    // relaxed, device scope -> global_atomic_add_f32 without return
    __hip_atomic_fetch_add(p, v, __ATOMIC_RELAXED, __HIP_MEMORY_SCOPE_AGENT);
}

__device__ __forceinline__ void async_load_b128_to_lds(const float* gsrc,
                                                       const float* lds_dst) {
    // Low 32 bits of a generic pointer into the LDS aperture are the
    // workgroup-relative LDS byte offset (ISA §10.2 aperture mapping).
    unsigned lds_off = (unsigned)(unsigned long long)lds_dst;
    unsigned long long ga = (unsigned long long)gsrc;
    asm volatile("global_load_async_to_lds_b128 %0, %1, off"
                 :: "v"(lds_off), "v"(ga)
                 : "memory");
}

__device__ __forceinline__ void wait_asynccnt0() {
#if defined(__has_builtin) && __has_builtin(__builtin_amdgcn_s_wait_asynccnt)
    __builtin_amdgcn_s_wait_asynccnt(0);
#else
    asm volatile("s_wait_asynccnt 0" ::: "memory");
#endif
}

// ---------------------------------------------------------------------------
// Transpose the three [512,256] weight matrices into [256,512] row-major so
// the WMMA B operand gather (per-lane contiguous float2 along k) is cheap.
// ---------------------------------------------------------------------------
__global__ __launch_bounds__(256) void transpose_w_kernel(
    const float* __restrict__ Wz, const float* __restrict__ Wr,
    const float* __restrict__ Wh,
    float* __restrict__ Tz, float* __restrict__ Tr, float* __restrict__ Th)
{
    int idx = blockIdx.x * blockDim.x + threadIdx.x;       // 0 .. 3*512*256-1
    const int per = DINC * HIDC;
    if (idx >= 3 * per) return;
    int which = idx / per;
    int r = idx - which * per;
    int k = r / HIDC;          // 0..511
    int n = r - k * HIDC;      // 0..255
    const float* S = (which == 0) ? Wz : (which == 1) ? Wr : Wh;
    float* D       = (which == 0) ? Tz : (which == 1) ? Tr : Th;
    D[(size_t)n * DINC + k] = S[(size_t)k * HIDC + n];
}

// ---------------------------------------------------------------------------
// Degree histogram over destinations (self-loops folded in as +1 later).
// ---------------------------------------------------------------------------
__global__ __launch_bounds__(256) void degree_kernel(
    const long long* __restrict__ edst, float* __restrict__ deg, int nE)
{
    int e = blockIdx.x * blockDim.x + threadIdx.x;
    if (e >= nE) return;
    atomic_add_f32(&deg[(int)edst[e]], 1.0f);
}

__global__ __launch_bounds__(256) void rsqrt_kernel(float* __restrict__ d, int n)
{
    int i = blockIdx.x * blockDim.x + threadIdx.x;
    if (i >= n) return;
    d[i] = rsqrtf(d[i] + 1.0f);   // +1 self-loop; always > 0
}

// ---------------------------------------------------------------------------
// WMMA GEMM:  out[N,256] = concat(Ax, Ah)[N,512] @ Wt^T   (Wt is [256,512]).
// One workgroup = one 16-row strip. 512 threads = 16 waves; wave w owns
// the 16-column tile [16*w, 16*w+16). A strip staged in LDS via async B128.
// Per k-step of 4: lane (m = lane&15, half = lane>>4) holds
//   A: A[m, kb + 2*half + {0,1}]      (ISA 32-bit A 16x4 layout)
//   B: Wt[n0+m, kb + 2*half + {0,1}]  (ISA 32-bit B 4x16 layout)
// C/D: VGPR v -> row v + 8*half, col = lane&15.
// ---------------------------------------------------------------------------
__global__ __launch_bounds__(512) void gemm_wmma_kernel(
    const float* __restrict__ Ax,   // [N,256] first half of concat
    const float* __restrict__ Ah,   // [N,256] second half of concat
    const float* __restrict__ Wt,   // [256,512] transposed weights
    float* __restrict__ out,        // [N,256]
    int nrows)
{
    __shared__ float lds[16 * LDS_STRIDE];

    const int row0 = blockIdx.x * 16;
    const int t = threadIdx.x;

    // Stage 16 x 512 fp32 strip of concat(Ax, Ah): 2048 async B128 copies,
    // 512 lanes x 4 rounds, memory -> LDS without touching VGPR data path.
#pragma unroll
    for (int j = 0; j < 4; ++j) {
        int idx4 = j * 512 + t;        // which 16-byte chunk
        int r = idx4 >> 7;             // row 0..15  (128 chunks per row)
        int k = (idx4 & 127) << 2;     // float col 0..508
        int rr = row0 + r;
        if (rr >= nrows) rr = nrows - 1;   // clamp (N=50000 is 16-divisible)
        const float* src = (k < HIDC) ? (Ax + (size_t)rr * HIDC + k)
                                      : (Ah + (size_t)rr * HIDC + (k - HIDC));
        async_load_b128_to_lds(src, &lds[r * LDS_STRIDE + k]);
    }
    wait_asynccnt0();
    __syncthreads();

    const int wave = t >> 5;          // 0..15 -> column tile
    const int lane = t & 31;
    const int m    = lane & 15;
    const int half = lane >> 4;       // 0 or 1
    const int col0 = wave * 16;
    const int kl   = 2 * half;        // per-lane k sub-offset

    const float* arow = &lds[m * LDS_STRIDE];                 // LDS, row m
    const float* brow = Wt + (size_t)(col0 + m) * DINC;       // global, L2-hot

    v8f acc = {};
#pragma unroll 4
    for (int kb = 0; kb < DINC; kb += 4) {
        v2f a = *(const v2f*)(arow + kb + kl);   // ds_load_b64
        v2f b = *(const v2f*)(brow + kb + kl);   // global_load_b64
        acc = __builtin_amdgcn_wmma_f32_16x16x4_f32(
            /*neg_a=*/false, a, /*neg_b=*/false, b,
            /*c_mod=*/(short)0, acc, /*reuse_a=*/false, /*reuse_b=*/false);
    }

    // Write 16x16 C tile; uniform bounds branch, no per-element exec churn.
    if (row0 + 16 <= nrows) {
#pragma unroll
        for (int v = 0; v < 8; ++v)
            out[(size_t)(row0 + v + 8 * half) * HIDC + col0 + m] = acc[v];
    } else {
#pragma unroll
        for (int v = 0; v < 8; ++v) {
            int r = row0 + v + 8 * half;
            if (r < nrows)
                out[(size_t)r * HIDC + col0 + m] = acc[v];
        }
    }
}

// ---------------------------------------------------------------------------
// Fused z/r scatter: one wave per edge. Reads two coalesced 1KB rows of
// XW[src], scales by norm = dinv[s]*dinv[d], scatter-adds into agg[dst].
// ---------------------------------------------------------------------------
__global__ __launch_bounds__(256) void scatter2_kernel(
    const long long* __restrict__ esrc, const long long* __restrict__ edst,
    const float* __restrict__ dinv,
    const float* __restrict__ XA, const float* __restrict__ XB,
    float* __restrict__ aggA, float* __restrict__ aggB, int nE)
{
    int e = blockIdx.x * (blockDim.x >> 5) + (threadIdx.x >> 5);
    if (e >= nE) return;
    int lane = threadIdx.x & 31;
    int s = (int)esrc[e];
    int d = (int)edst[e];
    float w = dinv[s] * dinv[d];

    size_t soff = (size_t)s * HIDC + lane * 8;
    size_t doff = (size_t)d * HIDC + lane * 8;
    float4 a0 = *(const float4*)(XA + soff);
    float4 a1 = *(const float4*)(XA + soff + 4);
    float4 b0 = *(const float4*)(XB + soff);
    float4 b1 = *(const float4*)(XB + soff + 4);
    float* pa = aggA + doff;
    float* pb = aggB + doff;
    atomic_add_f32(pa + 0, a0.x * w); atomic_add_f32(pa + 1, a0.y * w);
    atomic_add_f32(pa + 2, a0.z * w); atomic_add_f32(pa + 3, a0.w * w);
    atomic_add_f32(pa + 4, a1.x * w); atomic_add_f32(pa + 5, a1.y * w);
    atomic_add_f32(pa + 6, a1.z * w); atomic_add_f32(pa + 7, a1.w * w);
    atomic_add_f32(pb + 0, b0.x * w); atomic_add_f32(pb + 1, b0.y * w);
    atomic_add_f32(pb + 2, b0.z * w); atomic_add_f32(pb + 3, b0.w * w);
    atomic_add_f32(pb + 4, b1.x * w); atomic_add_f32(pb + 5, b1.y * w);
    atomic_add_f32(pb + 6, b1.z * w); atomic_add_f32(pb + 7, b1.w * w);
}

__global__ __launch_bounds__(256) void scatter1_kernel(
    const long long* __restrict__ esrc, const long long* __restrict__ edst,
    const float* __restrict__ dinv,
    const float* __restrict__ XA, float* __restrict__ aggA, int nE)
{
    int e = blockIdx.x * (blockDim.x >> 5) + (threadIdx.x >> 5);
    if (e >= nE) return;
    int lane = threadIdx.x & 31;
    int s = (int)esrc[e];
    int d = (int)edst[e];
    float w = dinv[s] * dinv[d];

    size_t soff = (size_t)s * HIDC + lane * 8;
    size_t doff = (size_t)d * HIDC + lane * 8;
    float4 a0 = *(const float4*)(XA + soff);
    float4 a1 = *(const float4*)(XA + soff + 4);
    float* pa = aggA + doff;
    atomic_add_f32(pa + 0, a0.x * w); atomic_add_f32(pa + 1, a0.y * w);
    atomic_add_f32(pa + 2, a0.z * w); atomic_add_f32(pa + 3, a0.w * w);
    atomic_add_f32(pa + 4, a1.x * w); atomic_add_f32(pa + 5, a1.y * w);
    atomic_add_f32(pa + 6, a1.z * w); atomic_add_f32(pa + 7, a1.w * w);
}

// ---------------------------------------------------------------------------
// Gate activations: fold in self-loop term XW[i]*dinv[i]^2 + bias, apply
// sigmoid; overwrite agg_z with z and agg_r with r*h (input to GEMM2).
// ---------------------------------------------------------------------------
__global__ __launch_bounds__(256) void act_zr_kernel(
    float* __restrict__ aggz, float* __restrict__ aggr,
    const float* __restrict__ xwz, const float* __restrict__ xwr,
    const float* __restrict__ h, const float* __restrict__ bz,
    const float* __restrict__ br, const float* __restrict__ dinv,
    long long total)
{
    long long idx = (long long)blockIdx.x * blockDim.x + threadIdx.x;
    if (idx >= total) return;
    int node = (int)(idx >> 8);
    int c = (int)(idx & 255);
    float di = dinv[node];
    float sl = di * di;
    float az = aggz[idx] + xwz[idx] * sl + bz[c];
    float ar = aggr[idx] + xwr[idx] * sl + br[c];
    float z = 1.0f / (1.0f + expf(-az));
    float r = 1.0f / (1.0f + expf(-ar));
    aggz[idx] = z;
    aggr[idx] = r * h[idx];
}

__global__ __launch_bounds__(256) void final_kernel(
    float* __restrict__ out, const float* __restrict__ aggh,
    const float* __restrict__ xwh, const float* __restrict__ zbuf,
    const float* __restrict__ h, const float* __restrict__ bh,
    const float* __restrict__ dinv, long long total)
{
    long long idx = (long long)blockIdx.x * blockDim.x + threadIdx.x;
    if (idx >= total) return;
    int node = (int)(idx >> 8);
    int c = (int)(idx & 255);
    float di = dinv[node];
    float sl = di * di;
    float ht = tanhf(aggh[idx] + xwh[idx] * sl + bh[c]);
    float z = zbuf[idx];
    out[idx] = z * h[idx] + (1.0f - z) * ht;
}

// ---------------------------------------------------------------------------
extern "C" void kernel_launch(void* const* d_in, const int* in_sizes, int n_in,
                              void* d_out, int out_size, void* d_ws, size_t ws_size,
                              hipStream_t stream)
{
    const float* x  = (const float*)d_in[0];
    const float* h  = (const float*)d_in[1];
    const long long* ei = (const long long*)d_in[2];   // int64 [2, E]
    const float* Wz = (const float*)d_in[3];
    const float* bz = (const float*)d_in[4];
    const float* Wr = (const float*)d_in[5];
    const float* br = (const float*)d_in[6];
    const float* Wh = (const float*)d_in[7];
    const float* bh = (const float*)d_in[8];
    float* out = (float*)d_out;

    const int N = in_sizes[0] / HIDC;
    const int E = in_sizes[2] / 2;
    const long long* esrc = ei;
    const long long* edst = ei + E;

    const size_t nh = (size_t)N * HIDC;
    float* buf0 = (float*)d_ws;        // XWz -> XWh
    float* buf1 = buf0 + nh;           // XWr -> agg_h
    float* buf2 = buf1 + nh;           // agg_z -> z
    float* buf3 = buf2 + nh;           // agg_r -> r*h
    float* dinv = buf3 + nh;           // deg -> dinv  (N floats)
    float* Tz = dinv + N;              // [256,512] each
    float* Tr = Tz + DINC * HIDC;
    float* Th = Tr + DINC * HIDC;

    const long long total = (long long)nh;
    const int strips = (N + 15) / 16;           // 3125 for N=50000
    const int ew_blocks = (E + 7) / 8;          // 8 waves (edges) per block
    const int el_blocks = (E + 255) / 256;
    const int nl_blocks = (N + 255) / 256;
    const int ch_blocks = (int)((total + 255) / 256);

    // Zero aggregation buffers + degree every call (graph-capture safe).
    hipMemsetAsync(buf2, 0, 2 * nh * sizeof(float), stream);  // agg_z, agg_r
    hipMemsetAsync(dinv, 0, (size_t)N * sizeof(float), stream);

    transpose_w_kernel<<<(3 * DINC * HIDC + 255) / 256, 256, 0, stream>>>(
        Wz, Wr, Wh, Tz, Tr, Th);
    degree_kernel<<<el_blocks, 256, 0, stream>>>(edst, dinv, E);
    rsqrt_kernel<<<nl_blocks, 256, 0, stream>>>(dinv, N);

    // z / r GEMMs: [x||h] @ W
    gemm_wmma_kernel<<<strips, 512, 0, stream>>>(x, h, Tz, buf0, N);
    gemm_wmma_kernel<<<strips, 512, 0, stream>>>(x, h, Tr, buf1, N);

    // Fused z/r edge aggregation.
    scatter2_kernel<<<ew_blocks, 256, 0, stream>>>(
        esrc, edst, dinv, buf0, buf1, buf2, buf3, E);

    // Gates: buf2 <- z, buf3 <- r*h.
    act_zr_kernel<<<ch_blocks, 256, 0, stream>>>(
        buf2, buf3, buf0, buf1, h, bz, br, dinv, total);

    // Candidate GEMM: [x || r*h] @ Wh  -> buf0 (XWz dead).
    gemm_wmma_kernel<<<strips, 512, 0, stream>>>(x, buf3, Th, buf0, N);

    // h aggregation into buf1 (XWr dead).
    hipMemsetAsync(buf1, 0, nh * sizeof(float), stream);
    scatter1_kernel<<<ew_blocks, 256, 0, stream>>>(
        esrc, edst, dinv, buf0, buf1, E);

    // out = z*h + (1-z)*tanh(agg_h + XWh*dinv^2 + bh)
    final_kernel<<<ch_blocks, 256, 0, stream>>>(
        out, buf1, buf0, buf2, h, bh, dinv, total);
}